// DetailPooling_63178968924231
// MI455X (gfx1250) — compile-verified
//
#include <hip/hip_runtime.h>
#include <hip/hip_bf16.h>

// DetailPooling on MI455X (gfx1250).
// Bandwidth-bound (320 MB traffic, ~13.7us floor @ 23.3 TB/s) -> optimize data
// movement, not FLOPs: async global->LDS tile staging (CDNA5 ASYNCcnt path),
// single-read of every input element, NT stores for the streaming output.

static constexpr int Himg  = 256;
static constexpr int Wimg  = 256;
static constexpr int OHimg = 128;
static constexpr int OWimg = 128;
static constexpr int TILE_O = 32;              // 32x32 outputs per workgroup
static constexpr int EQ_T   = 2 * TILE_O + 1;  // 65: eq4 tile (65x65)
static constexpr int XS_T   = EQ_T + 2;        // 67: x tile incl. blur halo
static constexpr int XS_P   = 68;              // padded LDS row stride (banks)
static constexpr int NTHREADS = 256;

typedef __attribute__((address_space(1))) int gint_t;   // global (AS1)
typedef __attribute__((address_space(3))) int sint_t;   // LDS (AS3)

__global__ __launch_bounds__(NTHREADS)
void detail_pool_kernel(const float* __restrict__ x,
                        const float* __restrict__ lam,
                        const float* __restrict__ alpha,
                        float* __restrict__ out)
{
    __shared__ float xs  [XS_T * XS_P];   // 67x68 f32 = 18.2 KB
    __shared__ float eq4s[EQ_T * XS_P];   // 65x68 f32 = 17.7 KB

    const int tid   = threadIdx.x;
    const int plane = blockIdx.y;                       // B*C planes
    const int tx    = blockIdx.x & 3;
    const int ty    = blockIdx.x >> 2;
    const int OH0   = ty * TILE_O;
    const int OW0   = tx * TILE_O;
    const int IH0   = 2 * OH0 - 1;                      // top-left incl. halo
    const int IW0   = 2 * OW0 - 1;

    const float* xp = x   + (size_t)plane * Himg * Wimg;
    float*       op = out + (size_t)plane * OHimg * OWimg;

    // ---- Phase 1: async-stage the 67x67 x tile (replicate-clamped) to LDS.
    // Per-lane clamped gather via global_load_async_to_lds_b32 (ASYNCcnt).
    for (int i = tid; i < XS_T * XS_T; i += NTHREADS) {
        int r = i / XS_T;
        int c = i - r * XS_T;
        int gh = IH0 + r; gh = gh < 0 ? 0 : (gh > Himg - 1 ? Himg - 1 : gh);
        int gw = IW0 + c; gw = gw < 0 ? 0 : (gw > Wimg - 1 ? Wimg - 1 : gw);
#if __has_builtin(__builtin_amdgcn_global_load_async_to_lds_b32)
        __builtin_amdgcn_global_load_async_to_lds_b32(
            (gint_t*)(xp + gh * Wimg + gw),
            (sint_t*)(xs + r * XS_P + c),
            /*offset=*/0, /*cpol=*/0);
#else
        xs[r * XS_P + c] = xp[gh * Wimg + gw];
#endif
    }
#if __has_builtin(__builtin_amdgcn_s_wait_asynccnt)
    __builtin_amdgcn_s_wait_asynccnt(0);
#else
    asm volatile("s_wait_asynccnt 0" ::: "memory");
#endif
    __syncthreads();

    // ---- Phase 2: eq4 = pow((blur_diff^2 + eps^2)^2, |lam|) + |alpha|,
    // computed ONCE per input pixel into LDS (each value reused by up to 4
    // outputs via the stride-1 2x2 denom pool).
    const float la2 = 2.0f * fabsf(lam[0]);   // (b^2)^la = exp2(2*la*log2 b)
    const float al  = fabsf(alpha[0]);
    for (int i = tid; i < EQ_T * EQ_T; i += NTHREADS) {
        int r = i / EQ_T;
        int c = i - r * EQ_T;
        // eq4 index 256 replicates 255 (edge pad of eq4, not recomputation):
        int rr = (2 * OH0 + r > Himg - 1) ? (Himg - 1 - 2 * OH0) : r;
        int cc = (2 * OW0 + c > Wimg - 1) ? (Wimg - 1 - 2 * OW0) : c;
        const float* p0 = xs + rr * XS_P + cc;
        float b0 = p0[0]          + 2.0f * p0[1]          + p0[2];
        float b1 = p0[XS_P]       + 2.0f * p0[XS_P + 1]   + p0[XS_P + 2];
        float b2 = p0[2 * XS_P]   + 2.0f * p0[2 * XS_P+1] + p0[2 * XS_P + 2];
        float blur = (b0 + 2.0f * b1 + b2) * (1.0f / 16.0f);
        float d    = p0[XS_P + 1] - blur;               // x - eq2
        float base = d * d + 1e-12f;                    // (x-eq2)^2 + eps^2
        // v_log_f32 is log2 on AMDGPU; (base^2)^la = exp2(2*la*log2(base))
        float eq56 = __builtin_amdgcn_exp2f(la2 * __builtin_amdgcn_logf(base));
        eq4s[r * XS_P + c] = eq56 + al;
    }
    __syncthreads();

    // ---- Phase 3: denom = avgpool2x2/s1(eq4,edge-pad)+1e-8 ;
    //               out   = avgpool2x2/s2( x * eq4/denom ).
    for (int i = tid; i < TILE_O * TILE_O; i += NTHREADS) {
        int orr = i >> 5;
        int occ = i & 31;
        float acc = 0.0f;
#pragma unroll
        for (int dy = 0; dy < 2; ++dy) {
#pragma unroll
            for (int dx = 0; dx < 2; ++dx) {
                int a = 2 * orr + dy;
                int b = 2 * occ + dx;
                const float* e0 = eq4s + a * XS_P + b;
                float e   = e0[0];
                float den = 0.25f * (e + e0[1] + e0[XS_P] + e0[XS_P + 1])
                          + 1e-8f;
                float xv  = xs[(a + 1) * XS_P + (b + 1)];
                acc += xv * e / den;
            }
        }
        // Streaming 64MB output: NT store keeps L2 for the input stream.
        __builtin_nontemporal_store(0.25f * acc,
                                    op + (OH0 + orr) * OWimg + (OW0 + occ));
    }
}

extern "C" void kernel_launch(void* const* d_in, const int* in_sizes, int n_in,
                              void* d_out, int out_size, void* d_ws, size_t ws_size,
                              hipStream_t stream) {
    const float* x     = (const float*)d_in[0];
    const float* lam   = (const float*)d_in[1];
    const float* alpha = (const float*)d_in[2];
    float* out = (float*)d_out;

    int planes = in_sizes[0] / (Himg * Wimg);           // B*C = 1024
    dim3 grid((OHimg / TILE_O) * (OWimg / TILE_O),      // 16 tiles/plane
              planes);
    detail_pool_kernel<<<grid, NTHREADS, 0, stream>>>(x, lam, alpha, out);
}